// GraphSAGE_51745765982698
// MI455X (gfx1250) — compile-verified
//
#include <hip/hip_runtime.h>
#include <hip/hip_bf16.h>

#define D 64   // D_IN == D_OUT == 64

typedef __attribute__((ext_vector_type(2))) float v2f;
typedef __attribute__((ext_vector_type(8))) float v8f;

// ---------------------------------------------------------------------------
// Kernel 1: zero workspace (neigh_sum [N*64] followed by deg [N])
// ---------------------------------------------------------------------------
__global__ void sage_zero_kernel(float* __restrict__ p, int n) {
    int i = blockIdx.x * blockDim.x + threadIdx.x;
    int stride = gridDim.x * blockDim.x;
    for (; i < n; i += stride) p[i] = 0.0f;
}

// ---------------------------------------------------------------------------
// Kernel 2: edge aggregation. One wave32 per edge.
// Lane l handles feature elements 2l, 2l+1 (float2, coalesced 256B per wave).
// neigh_sum[dst] += h[src]; deg[dst] += 1  (f32 global atomics, resolve in L2)
// ---------------------------------------------------------------------------
__global__ void sage_edge_agg_kernel(const float* __restrict__ h,
                                     const int* __restrict__ src,
                                     const int* __restrict__ dst,
                                     float* __restrict__ nsum,
                                     float* __restrict__ deg,
                                     int nEdges) {
    int gtid = blockIdx.x * blockDim.x + threadIdx.x;
    int edge = gtid >> 5;
    int lane = threadIdx.x & 31;
    if (edge >= nEdges) return;

    int s = src[edge];
    int d = dst[edge];

    const float2* hp = (const float2*)(h + (size_t)s * D);
    float2 v = hp[lane];

    float* np = nsum + (size_t)d * D + lane * 2;
    atomicAdd(np,     v.x);
    atomicAdd(np + 1, v.y);
    if (lane == 0) atomicAdd(deg + d, 1.0f);
}

// ---------------------------------------------------------------------------
// Kernel 3: fused GEMM  out = (nsum/deg) @ Wn + h @ Ws  using f32 WMMA.
//
// B fragments are PRE-PACKED into LDS in the exact V_WMMA_F32_16X16X4_F32
// register layout: fragment f = n*16 + i (n = 16-col tile, i = K/4 step),
// lane l needs {W[kk][c], W[kk+1][c]} with kk = 4i + 2*(l>=16), c = n*16+(l&15).
// Stored at sB[f*32 + l] so the inner loop is one lane-linear ds_load_b64
// (conflict-free, constant immediate offset) feeding WMMA directly.
//
// 256 threads = 8 waves; one wave per 16-node tile. N = 100000 = 16*6250 so
// all active waves have EXEC all-1s (WMMA requirement); tail waves exit via a
// wave-uniform branch.
// ---------------------------------------------------------------------------
__global__ void __launch_bounds__(256)
sage_wmma_gemm_kernel(const float* __restrict__ nsum,
                      const float* __restrict__ deg,
                      const float* __restrict__ h,
                      const float* __restrict__ Wn,
                      const float* __restrict__ Ws,
                      float* __restrict__ out,
                      int nNodes) {
    __shared__ v2f sBn[64 * 32];   // 16 KB: pre-packed W_neigh fragments
    __shared__ v2f sBs[64 * 32];   // 16 KB: pre-packed W_self  fragments

    // Cooperative staging: build fragments in WMMA lane layout.
    for (int idx = threadIdx.x; idx < 64 * 32; idx += blockDim.x) {
        int f = idx >> 5;                    // fragment 0..63
        int l = idx & 31;                    // lane slot
        int n = f >> 4;                      // output col tile 0..3
        int i = f & 15;                      // K step 0..15
        int kk = 4 * i + ((l >> 4) << 1);    // K base for this lane half
        int c  = n * 16 + (l & 15);          // output column
        v2f bn, bs;
        bn.x = Wn[kk * D + c];
        bn.y = Wn[(kk + 1) * D + c];
        bs.x = Ws[kk * D + c];
        bs.y = Ws[(kk + 1) * D + c];
        sBn[idx] = bn;
        sBs[idx] = bs;
    }
    __syncthreads();

    int wave = threadIdx.x >> 5;                 // 0..7
    int lane = threadIdx.x & 31;
    int tile = blockIdx.x * 8 + wave;            // 16-row node tile
    int row0 = tile * 16;
    if (row0 >= nNodes) return;                  // wave-uniform: EXEC stays full

    int row   = lane & 15;                       // M for A, N for B/C/D
    int khalf = (lane >> 4) << 1;                // lanes 16-31 own K+2,K+3
    int r     = row0 + row;                      // global node row (in range)

    float dv  = deg[r];
    float inv = 1.0f / fmaxf(dv, 1.0f);

    // Preload this wave's A-rows: 16 K-chunks of float2 per matrix.
    const float* nrow = nsum + (size_t)r * D;
    const float* hrow = h    + (size_t)r * D;
    v2f an[16], ah[16];
#pragma unroll
    for (int i = 0; i < 16; ++i) {
        int kk = 4 * i + khalf;
        v2f a;
        a.x = nrow[kk] * inv;
        a.y = nrow[kk + 1] * inv;
        an[i] = a;
        v2f b;
        b.x = hrow[kk];
        b.y = hrow[kk + 1];
        ah[i] = b;
    }

    const v2f* bnp = sBn + lane;                 // lane-linear base, offset f*32
    const v2f* bsp = sBs + lane;

#pragma unroll
    for (int n = 0; n < 4; ++n) {                // four 16-col output tiles
        v8f acc = {};
#pragma unroll
        for (int i = 0; i < 16; ++i) {           // K = 64 in steps of 4
            int f = n * 16 + i;
            v2f bn = bnp[f * 32];                // ds_load_b64, imm offset
            v2f bs = bsp[f * 32];
            acc = __builtin_amdgcn_wmma_f32_16x16x4_f32(
                false, an[i], false, bn, (short)0, acc, false, false);
            acc = __builtin_amdgcn_wmma_f32_16x16x4_f32(
                false, ah[i], false, bs, (short)0, acc, false, false);
        }
        // C/D layout: VGPR j -> M = j + 8*(lane>=16); N = lane&15
        int mbase = row0 + ((lane >> 4) << 3);
#pragma unroll
        for (int j = 0; j < 8; ++j) {
            out[(size_t)(mbase + j) * D + n * 16 + row] = acc[j];
        }
    }
}

// ---------------------------------------------------------------------------
// Launch: inputs are {h, W_neigh, W_self, src, dst}; output f32 [N, 64].
// Workspace layout: nsum = N*64 floats, then deg = N floats (26 MB total).
// ---------------------------------------------------------------------------
extern "C" void kernel_launch(void* const* d_in, const int* in_sizes, int n_in,
                              void* d_out, int out_size, void* d_ws, size_t ws_size,
                              hipStream_t stream) {
    const float* h  = (const float*)d_in[0];
    const float* Wn = (const float*)d_in[1];
    const float* Ws = (const float*)d_in[2];
    const int* src  = (const int*)d_in[3];
    const int* dst  = (const int*)d_in[4];
    float* out      = (float*)d_out;

    int nNodes = in_sizes[0] / D;
    int nEdges = in_sizes[3];

    float* nsum = (float*)d_ws;
    float* deg  = nsum + (size_t)nNodes * D;

    // 1) zero accumulators
    int zn = nNodes * (D + 1);
    sage_zero_kernel<<<2048, 256, 0, stream>>>(nsum, zn);

    // 2) edge gather + atomic scatter-add (one wave32 per edge)
    int aggBlocks = (nEdges + 7) / 8;            // 8 edges per 256-thread block
    sage_edge_agg_kernel<<<aggBlocks, 256, 0, stream>>>(h, src, dst, nsum, deg, nEdges);

    // 3) fused mean + dual-GEMM with f32 WMMA (one wave per 16-node tile)
    int tiles = (nNodes + 15) / 16;
    int gemmBlocks = (tiles + 7) / 8;
    sage_wmma_gemm_kernel<<<gemmBlocks, 256, 0, stream>>>(nsum, deg, h, Wn, Ws, out, nNodes);
}